// VelocityField_5927054868764
// MI455X (gfx1250) — compile-verified
//
#include <hip/hip_runtime.h>
#include <math.h>

typedef float v2f __attribute__((ext_vector_type(2)));
typedef float v8f __attribute__((ext_vector_type(8)));
typedef int   v4i __attribute__((vector_size(16)));   // matches builtin param pointee

#define WAVES_PER_BLOCK 4
#define BLOCK_THREADS   128
#define NBLOCKS         512
#define TILE_ROW        68            // padded row stride (floats); 64 data + 4 pad
#define TILE_FLOATS     (16 * TILE_ROW)

#if __has_builtin(__builtin_amdgcn_global_load_async_to_lds_b128) && \
    __has_builtin(__builtin_amdgcn_s_wait_asynccnt)
#define USE_ASYNC 1
#else
#define USE_ASYNC 0
#endif

__device__ __forceinline__ void async_copy16(const float* gsrc, float* ldst)
{
#if USE_ASYNC
    __builtin_amdgcn_global_load_async_to_lds_b128(
        (__attribute__((address_space(1))) v4i*)gsrc,
        (__attribute__((address_space(3))) v4i*)ldst,
        /*offset=*/0, /*cpol=*/0);
#else
    (void)gsrc; (void)ldst;
#endif
}

// ---------------------------------------------------------------------------
// Pass 1: stream X once; per-block partial sums of [ T_0..T_63 , SUMW ].
// Each wave processes 16-sample x 64-dim tiles, double-buffered in LDS via
// async global->LDS copies, with f32 WMMA computing dot(r, x_n).
// ---------------------------------------------------------------------------
__global__ __launch_bounds__(BLOCK_THREADS)
void vf_partial_kernel(const float* __restrict__ r,
                       const float* __restrict__ X,
                       float* __restrict__ partials,   // [NBLOCKS][65]
                       int nsamp)
{
    __shared__ float tile[WAVES_PER_BLOCK * 2 * TILE_FLOATS];
    __shared__ float red[WAVES_PER_BLOCK * 65];

    const int tid  = threadIdx.x;
    const int wv   = tid >> 5;
    const int lane = tid & 31;
    const int half = lane >> 4;        // 0: K={0,1}, 1: K={2,3}
    const int n    = lane & 15;        // sample column within tile

    float* const buf0 = &tile[(wv * 2 + 0) * TILE_FLOATS];
    float* const buf1 = &tile[(wv * 2 + 1) * TILE_FLOATS];

    // Build replicated-A (r chunks) in WMMA 16x16x4 f32 A layout.
    // Every row m holds the same r chunk, so every row of D equals r.x_n.
    v2f A[16];
    float rrp = 0.f;
#pragma unroll
    for (int c = 0; c < 16; ++c) {
        const float2 rv = *reinterpret_cast<const float2*>(r + 4 * c + 2 * half);
        A[c].x = rv.x; A[c].y = rv.y;
        rrp += rv.x * rv.x + rv.y * rv.y;
    }
    const float rr = rrp + __shfl_xor(rrp, 16, 32);   // ||r||^2, all lanes

    float acc0 = 0.f, acc1 = 0.f, accS0 = 0.f;

    const int ntiles = (nsamp + 15) >> 4;
    const int gwave  = blockIdx.x * WAVES_PER_BLOCK + wv;
    const int stride = gridDim.x * WAVES_PER_BLOCK;

    // Issue tile t's 16x64 f32 block into `buf`.  Returns true if issued as
    // async global->LDS copies (full tile only), false if synchronous path.
    auto issue_tile = [&](int t, float* buf) -> bool {
        const long i0 = (long)t * 16;
#if USE_ASYNC
        if (i0 + 16 <= (long)nsamp) {
#pragma unroll
            for (int it = 0; it < 8; ++it) {
                const int f  = it * 32 + lane;        // float4 chunk id 0..255
                const int s  = f >> 4;
                const int c4 = (f & 15) << 2;
                async_copy16(X + (i0 + s) * 64 + c4, &buf[s * TILE_ROW + c4]);
            }
            return true;
        }
#endif
#pragma unroll
        for (int it = 0; it < 8; ++it) {
            const int f  = it * 32 + lane;
            const int s  = f >> 4;
            const int c4 = (f & 15) << 2;
            float4 v = make_float4(0.f, 0.f, 0.f, 0.f);
            if (i0 + s < nsamp)
                v = *reinterpret_cast<const float4*>(X + (i0 + s) * 64 + c4);
            *reinterpret_cast<float4*>(&buf[s * TILE_ROW + c4]) = v;
        }
        return false;
    };

    int  t        = gwave;
    int  cur      = 0;
    bool curAsync = false;
    if (t < ntiles) curAsync = issue_tile(t, buf0);

    for (; t < ntiles; t += stride) {
        const int tn = t + stride;
        bool nextAsync = false;
        if (tn < ntiles) nextAsync = issue_tile(tn, cur ? buf0 : buf1);

#if USE_ASYNC
        if (curAsync) {
            // Async loads complete in order: <=8 outstanding means the
            // current buffer's 8 copies have landed in LDS.
            if (nextAsync) __builtin_amdgcn_s_wait_asynccnt(8);
            else           __builtin_amdgcn_s_wait_asynccnt(0);
            __asm__ volatile("" ::: "memory");       // keep ds_loads after wait
        }
#endif
        const float* mytile = cur ? buf1 : buf0;

        // ---- WMMA chain: dot(r, x_n); VALU alongside: ||x_n||^2 ----
        v8f D = {0.f, 0.f, 0.f, 0.f, 0.f, 0.f, 0.f, 0.f};
        float xxp = 0.f;
#pragma unroll
        for (int c = 0; c < 16; ++c) {
            const int k0 = 4 * c;
            // B (4x16) layout mirror of A: lane n holds rows K=2*half+{0,1}
            // of sample column n.  Stride-68 rows -> conflict-free b64 loads.
            const float2 bv = *reinterpret_cast<const float2*>(
                &mytile[n * TILE_ROW + k0 + 2 * half]);
            v2f B; B.x = bv.x; B.y = bv.y;
            xxp += bv.x * bv.x + bv.y * bv.y;
            D = __builtin_amdgcn_wmma_f32_16x16x4_f32(
                    false, A[c], false, B, (short)0, D, false, false);
        }
        const float xx  = xxp + __shfl_xor(xxp, 16, 32);
        const float dot = D[0];            // rows M=0 / M=8: identical by construction
        const float sq  = rr + xx - 2.f * dot;
        float w = expf(-2.f * sq);         // h = 0.5 -> 1/(2h^2) = 2
        if ((long)t * 16 + n >= (long)nsamp) w = 0.f;  // zero padded samples
        accS0 += w;                        // counted twice (both halves); halved later

        // ---- T_d += w_s * x_{s,d}: lane L owns dims L and L+32 ----
#pragma unroll
        for (int s = 0; s < 16; ++s) {
            const float ws = __shfl(w, s, 32);         // broadcast from lane s
            acc0 += ws * mytile[s * TILE_ROW + lane];
            acc1 += ws * mytile[s * TILE_ROW + 32 + lane];
        }

        cur ^= 1;
        curAsync = nextAsync;
    }

    // Intra-wave SUMW reduce (fixed tree), halve the double count.
    float s0 = accS0;
#pragma unroll
    for (int off = 16; off >= 1; off >>= 1) s0 += __shfl_xor(s0, off, 32);
    s0 *= 0.5f;

    red[wv * 65 + lane]      = acc0;
    red[wv * 65 + 32 + lane] = acc1;
    if (lane == 0) red[wv * 65 + 64] = s0;
    __syncthreads();

    if (tid < 65) {
        float a = 0.f;
        for (int w8 = 0; w8 < WAVES_PER_BLOCK; ++w8)   // fixed order: deterministic
            a += red[w8 * 65 + tid];
        partials[blockIdx.x * 65 + tid] = a;
    }
}

// ---------------------------------------------------------------------------
// Pass 2: fixed-order combine of block partials + finalize in f64.
// out_d = (r_d - T_d/SUMW)/h^2 - r_d * exp(-||r||^2/2) * N / (2^64 * SUMW)
// ---------------------------------------------------------------------------
__global__ __launch_bounds__(128)
void vf_finalize_kernel(const float* __restrict__ r,
                        const float* __restrict__ partials,
                        float* __restrict__ out,
                        int nblocks, int nsamp)
{
    __shared__ double sums[65];
    __shared__ double rr_sh;
    const int t = threadIdx.x;
    if (t < 65) {
        double a = 0.0;
        for (int b = 0; b < nblocks; ++b)              // fixed order: deterministic
            a += (double)partials[b * 65 + t];
        sums[t] = a;
    }
    __syncthreads();
    if (t == 0) {
        double rr = 0.0;
        for (int d = 0; d < 64; ++d) { const double rd = (double)r[d]; rr += rd * rd; }
        rr_sh = rr;
    }
    __syncthreads();
    if (t < 64) {
        const double SUMW = sums[64];
        const double Td   = sums[t];
        const double rd   = (double)r[t];
        const double C    = exp(-0.5 * rr_sh) * (double)nsamp
                            / (18446744073709551616.0 * SUMW);   // 2^64
        out[t] = (float)((rd - Td / SUMW) * 4.0 - rd * C);       // 1/h^2 = 4
    }
}

extern "C" void kernel_launch(void* const* d_in, const int* in_sizes, int n_in,
                              void* d_out, int out_size, void* d_ws, size_t ws_size,
                              hipStream_t stream)
{
    (void)n_in; (void)out_size; (void)ws_size;
    const float* r = (const float*)d_in[0];
    const float* X = (const float*)d_in[1];
    const int dim   = in_sizes[0];           // 64
    const int nsamp = in_sizes[1] / dim;     // 500000

    float* partials = (float*)d_ws;          // NBLOCKS * 65 floats = 130 KB

    vf_partial_kernel<<<NBLOCKS, BLOCK_THREADS, 0, stream>>>(r, X, partials, nsamp);
    vf_finalize_kernel<<<1, 128, 0, stream>>>(r, partials, (float*)d_out,
                                              NBLOCKS, nsamp);
}